// GNNLayer_7473243095220
// MI455X (gfx1250) — compile-verified
//
#include <hip/hip_runtime.h>
#include <hip/hip_bf16.h>

typedef __attribute__((ext_vector_type(2))) float v2f;
typedef __attribute__((ext_vector_type(8))) float v8f;

#define NEG_SLOPE 0.2f
#define BN_EPS 1e-5f

static __device__ __forceinline__ float atomAddF(float* p, float v) {
    return unsafeAtomicAdd(p, v);   // lowers to global_atomic_add_f32
}

// monotonic unsigned encoding of float for atomicMax
static __device__ __forceinline__ unsigned encodeF(float f) {
    unsigned u = __float_as_uint(f);
    return (u & 0x80000000u) ? ~u : (u | 0x80000000u);
}
static __device__ __forceinline__ float decodeF(unsigned e) {
    return (e & 0x80000000u) ? __uint_as_float(e & 0x7fffffffu)
                             : __uint_as_float(~e);
}

// ---------------------------------------------------------------------------
// 0) init: out = bias (broadcast), amax = 0 (== encoded -NaN floor), denom = 0,
//    sums/sumsq = 0. Must run every call (graph replay determinism).
// ---------------------------------------------------------------------------
__global__ void k_init(float* __restrict__ out, const float* __restrict__ bias,
                       unsigned* __restrict__ amax, float* __restrict__ denom,
                       float* __restrict__ sums, float* __restrict__ sumsq, int n) {
    int idx = blockIdx.x * blockDim.x + threadIdx.x;
    int total = n * 64;
    if (idx < total) out[idx] = bias[idx & 63];
    if (idx < n) { amax[idx] = 0u; denom[idx] = 0.0f; }
    if (idx < 64) { sums[idx] = 0.0f; sumsq[idx] = 0.0f; }
}

// ---------------------------------------------------------------------------
// 1) x = batch_mat @ W.T  via V_WMMA_F32_16X16X4_F32, wave32.
//    Each wave: one 16-row block x 64 cols (4 accumulators), K=64 in steps of 4.
// ---------------------------------------------------------------------------
__global__ void __launch_bounds__(256)
k_gemm(const float* __restrict__ A,   // [n,64] batch_mat
       const float* __restrict__ W,   // [64,64] (row = out channel)
       float* __restrict__ X, int n) {
    int wave = threadIdx.x >> 5;
    int lane = threadIdx.x & 31;
    int m0 = (blockIdx.x * 8 + wave) * 16;
    if (m0 >= n) return;
    int l15   = lane & 15;
    int khalf = lane >> 4;                  // 0 or 1
    int mrow  = m0 + l15;

    v8f acc[4] = {v8f{}, v8f{}, v8f{}, v8f{}};
    #pragma unroll
    for (int k = 0; k < 64; k += 4) {
        // A fragment: A[mrow][k + 2*khalf + {0,1}]
        v2f a = *(const v2f*)(A + (size_t)mrow * 64 + k + 2 * khalf);
        #pragma unroll
        for (int t = 0; t < 4; ++t) {
            int ncol = t * 16 + l15;
            // B fragment: B[kk][n] = W[n][kk] -> consecutive in W
            v2f b = *(const v2f*)(W + (size_t)ncol * 64 + k + 2 * khalf);
            acc[t] = __builtin_amdgcn_wmma_f32_16x16x4_f32(
                false, a, false, b, (short)0, acc[t], false, false);
        }
    }
    // C layout: VGPR r -> M = m0 + r + 8*khalf, N = t*16 + l15
    #pragma unroll
    for (int t = 0; t < 4; ++t) {
        #pragma unroll
        for (int r = 0; r < 8; ++r) {
            X[(size_t)(m0 + r + 8 * khalf) * 64 + t * 16 + l15] = acc[t][r];
        }
    }
}

// ---------------------------------------------------------------------------
// 2) per-node scores: s_i = x.att_i + emb.att_em_i ; s_j likewise
// ---------------------------------------------------------------------------
__global__ void k_scores(const float* __restrict__ X, const float* __restrict__ Emb,
                         const float* __restrict__ ai, const float* __restrict__ aj,
                         const float* __restrict__ aei, const float* __restrict__ aej,
                         float* __restrict__ s_i, float* __restrict__ s_j, int n) {
    int node = blockIdx.x * blockDim.x + threadIdx.x;
    if (node >= n) return;
    const float4* xr = (const float4*)(X + (size_t)node * 64);
    const float4* er = (const float4*)(Emb + (size_t)node * 64);
    float si = 0.0f, sj = 0.0f;
    #pragma unroll
    for (int q = 0; q < 16; ++q) {
        float4 xv = xr[q], ev = er[q];
        float4 vi = ((const float4*)ai)[q];
        float4 vj = ((const float4*)aj)[q];
        float4 wi = ((const float4*)aei)[q];
        float4 wj = ((const float4*)aej)[q];
        si += xv.x*vi.x + xv.y*vi.y + xv.z*vi.z + xv.w*vi.w
            + ev.x*wi.x + ev.y*wi.y + ev.z*wi.z + ev.w*wi.w;
        sj += xv.x*vj.x + xv.y*vj.y + xv.z*vj.z + xv.w*vj.w
            + ev.x*wj.x + ev.y*wj.y + ev.z*wj.z + ev.w*wj.w;
    }
    s_i[node] = si;
    s_j[node] = sj;
}

static __device__ __forceinline__ bool edge_sd(int e, int E, int n,
                                               const int* src, const int* dst,
                                               int& s, int& d) {
    if (e < E) {
        s = src[e]; d = dst[e];
        return s != d;                 // self-loops in list are masked out
    }
    s = d = e - E;                     // appended self-loop
    return true;
}

// ---------------------------------------------------------------------------
// 3) segment max of leaky_relu(s_i[d] + s_j[s]) over dst
// ---------------------------------------------------------------------------
__global__ void k_edge_max(const int* __restrict__ src, const int* __restrict__ dst,
                           const float* __restrict__ s_i, const float* __restrict__ s_j,
                           unsigned* __restrict__ amax, int E, int n) {
    int e = blockIdx.x * blockDim.x + threadIdx.x;
    if (e >= E + n) return;
    int s, d;
    if (!edge_sd(e, E, n, src, dst, s, d)) return;
    float a = s_i[d] + s_j[s];
    a = (a >= 0.0f) ? a : NEG_SLOPE * a;
    atomicMax(&amax[d], encodeF(a));
}

// ---------------------------------------------------------------------------
// 4) denom[d] += exp(alpha - amax[d])
// ---------------------------------------------------------------------------
__global__ void k_edge_denom(const int* __restrict__ src, const int* __restrict__ dst,
                             const float* __restrict__ s_i, const float* __restrict__ s_j,
                             const unsigned* __restrict__ amax, float* __restrict__ denom,
                             int E, int n) {
    int e = blockIdx.x * blockDim.x + threadIdx.x;
    if (e >= E + n) return;
    int s, d;
    if (!edge_sd(e, E, n, src, dst, s, d)) return;
    float a = s_i[d] + s_j[s];
    a = (a >= 0.0f) ? a : NEG_SLOPE * a;
    atomAddF(&denom[d], __expf(a - decodeF(amax[d])));
}

// ---------------------------------------------------------------------------
// 5) out[d] += x[s] * attn  (one wave32 per edge, 2 channels per lane)
// ---------------------------------------------------------------------------
__global__ void __launch_bounds__(256)
k_edge_agg(const int* __restrict__ src, const int* __restrict__ dst,
           const float* __restrict__ s_i, const float* __restrict__ s_j,
           const unsigned* __restrict__ amax, const float* __restrict__ denom,
           const float* __restrict__ X, float* __restrict__ out, int E, int n) {
    int gid = blockIdx.x * blockDim.x + threadIdx.x;
    int e = gid >> 5;
    int lane = gid & 31;
    if (e >= E + n) return;
    int s, d;
    if (!edge_sd(e, E, n, src, dst, s, d)) return;
    float a = s_i[d] + s_j[s];
    a = (a >= 0.0f) ? a : NEG_SLOPE * a;
    float attn = __expf(a - decodeF(amax[d])) / (denom[d] + 1e-16f);
    float2 xv = ((const float2*)(X + (size_t)s * 64))[lane];
    float* o = out + (size_t)d * 64 + 2 * lane;
    atomAddF(o + 0, xv.x * attn);
    atomAddF(o + 1, xv.y * attn);
}

// ---------------------------------------------------------------------------
// 6) BN stats: per-channel sum / sumsq (LDS partial + global f32 atomics)
// ---------------------------------------------------------------------------
__global__ void __launch_bounds__(256)
k_bn_stats(const float* __restrict__ out, float* __restrict__ sums,
           float* __restrict__ sumsq, int n) {
    __shared__ float ls[256], ls2[256];
    int c = threadIdx.x & 63;
    int rg = threadIdx.x >> 6;                     // 0..3
    float s = 0.0f, s2 = 0.0f;
    for (int row = blockIdx.x * 4 + rg; row < n; row += gridDim.x * 4) {
        float v = out[(size_t)row * 64 + c];
        s += v; s2 += v * v;
    }
    ls[threadIdx.x] = s; ls2[threadIdx.x] = s2;
    __syncthreads();
    if (threadIdx.x < 64) {
        s  = ls[threadIdx.x]  + ls[threadIdx.x + 64]  + ls[threadIdx.x + 128]  + ls[threadIdx.x + 192];
        s2 = ls2[threadIdx.x] + ls2[threadIdx.x + 64] + ls2[threadIdx.x + 128] + ls2[threadIdx.x + 192];
        atomAddF(&sums[threadIdx.x], s);
        atomAddF(&sumsq[threadIdx.x], s2);
    }
}

// ---------------------------------------------------------------------------
// 7) BN apply + ReLU (in place on out)
// ---------------------------------------------------------------------------
__global__ void k_bn_apply(float* __restrict__ out, const float* __restrict__ sums,
                           const float* __restrict__ sumsq, const float* __restrict__ gamma,
                           const float* __restrict__ beta, int n) {
    int idx = blockIdx.x * blockDim.x + threadIdx.x;
    if (idx >= n * 64) return;
    int c = idx & 63;
    float inv_n = 1.0f / (float)n;
    float mu  = sums[c] * inv_n;
    float var = sumsq[c] * inv_n - mu * mu;
    float v = (out[idx] - mu) * __frsqrt_rn(var + BN_EPS) * gamma[c] + beta[c];
    out[idx] = fmaxf(v, 0.0f);
}

extern "C" void kernel_launch(void* const* d_in, const int* in_sizes, int n_in,
                              void* d_out, int out_size, void* d_ws, size_t ws_size,
                              hipStream_t stream) {
    const float* batch = (const float*)d_in[0];
    const float* emb   = (const float*)d_in[1];
    const float* W     = (const float*)d_in[2];
    const float* att_i = (const float*)d_in[3];
    const float* att_j = (const float*)d_in[4];
    const float* aem_i = (const float*)d_in[5];
    const float* aem_j = (const float*)d_in[6];
    const float* bias  = (const float*)d_in[7];
    const float* gamma = (const float*)d_in[8];
    const float* beta  = (const float*)d_in[9];
    const int*   edges = (const int*)d_in[10];

    int n = in_sizes[0] / 64;
    int E = in_sizes[10] / 2;
    const int* src = edges;
    const int* dst = edges + E;
    float* out = (float*)d_out;

    // workspace layout
    float*    x     = (float*)d_ws;
    float*    s_i   = x + (size_t)n * 64;
    float*    s_j   = s_i + n;
    unsigned* amax  = (unsigned*)(s_j + n);
    float*    denom = (float*)(amax + n);
    float*    sums  = denom + n;
    float*    sumsq = sums + 64;

    int total   = n * 64;
    int EN      = E + n;

    k_init<<<(total + 255) / 256, 256, 0, stream>>>(out, bias, amax, denom, sums, sumsq, n);
    k_gemm<<<(n + 127) / 128, 256, 0, stream>>>(batch, W, x, n);
    k_scores<<<(n + 255) / 256, 256, 0, stream>>>(x, emb, att_i, att_j, aem_i, aem_j, s_i, s_j, n);
    k_edge_max<<<(EN + 255) / 256, 256, 0, stream>>>(src, dst, s_i, s_j, amax, E, n);
    k_edge_denom<<<(EN + 255) / 256, 256, 0, stream>>>(src, dst, s_i, s_j, amax, denom, E, n);
    k_edge_agg<<<(EN + 7) / 8, 256, 0, stream>>>(src, dst, s_i, s_j, amax, denom, x, out, E, n);
    k_bn_stats<<<256, 256, 0, stream>>>(out, sums, sumsq, n);
    k_bn_apply<<<(total + 255) / 256, 256, 0, stream>>>(out, sums, sumsq, gamma, beta, n);
}